// QuantumFeatureExtractor_83863531421723
// MI455X (gfx1250) — compile-verified
//
#include <hip/hip_runtime.h>

// ---------------------------------------------------------------------------
// 12-qubit, 4-layer variational circuit simulator for MI455X (gfx1250).
// One workgroup (8 wave32) per batch element; state lives in LDS.
// All state updates are fused 3-qubit 8x8 complex unitaries applied via
// V_WMMA_F32_16X16X4_F32 on the complex->real 16x16 embedding.
// v2: all 16 per-layer unitaries are built up-front into an 8 KB LDS bank,
// halving the workgroup-barrier count (one barrier per application).
// ---------------------------------------------------------------------------

typedef __attribute__((ext_vector_type(2))) float v2f;
typedef __attribute__((ext_vector_type(8))) float v8f;

#define NQ      12
#define DIM     4096        // 2^12 amplitudes
#define NPARAM  240         // 12*4*5
#define NTHREAD 256
#define NMAT    16          // matrices per layer: 4 rot + 6 fwd CRX + 6 bwd CRX

// CRX block tables. Triples are (qh<qm<ql); qh maps to bit2 (MSB) of the
// 3-bit group index, matching state axis order (qubit 0 = most significant).
static __device__ const int c_fwd_q[6][3] = {
    {0,1,2},{2,3,4},{4,5,6},{6,7,8},{8,9,10},{0,10,11}};
static __device__ const int c_fwd_p[6][4] = {   // {pc1,pt1,pc2,pt2}
    {2,1,1,0},{2,1,1,0},{2,1,1,0},{2,1,1,0},{2,1,1,0},{1,0,0,2}};
static __device__ const int c_bwd_q[6][3] = {
    {9,10,11},{7,8,9},{5,6,7},{3,4,5},{1,2,3},{0,1,11}};
static __device__ const int c_bwd_p[6][4] = {
    {0,1,1,2},{0,1,1,2},{0,1,1,2},{0,1,1,2},{0,1,1,2},{1,2,2,0}};

__device__ __forceinline__ float2 cmulf(float2 a, float2 b) {
    return make_float2(a.x * b.x - a.y * b.y, a.x * b.y + a.y * b.x);
}

// Entry (i,j) of a CRX gate embedded in an 8x8 (3-qubit) space.
// pc/pt = bit positions of control/target. RX: [[c,-is],[-is,c]].
__device__ __forceinline__ float2 crx_entry(int i, int j, int pc, int pt,
                                            float c, float s) {
    int ps = 3 - pc - pt;                       // spectator position
    if (((i ^ j) >> ps) & 1) return make_float2(0.f, 0.f);
    if (((i ^ j) >> pc) & 1) return make_float2(0.f, 0.f);
    int bc = (i >> pc) & 1;
    int td = ((i ^ j) >> pt) & 1;
    if (!bc) return td ? make_float2(0.f, 0.f) : make_float2(1.f, 0.f);
    return td ? make_float2(0.f, -s) : make_float2(c, 0.f);
}

// Kronecker entry (i,j) of Uh (x) Um (x) Ul (MSB qubit first).
__device__ __forceinline__ float2 kron_entry(const float2* __restrict__ U,
                                             int qh, int qm, int ql,
                                             int i, int j) {
    float2 a = U[4 * qh + (((i >> 2) & 1) << 1) + ((j >> 2) & 1)];
    float2 b = U[4 * qm + (((i >> 1) & 1) << 1) + ((j >> 1) & 1)];
    float2 c = U[4 * ql + ((i & 1) << 1) + (j & 1)];
    return cmulf(cmulf(a, b), c);
}

// M = G2 * G1 entry (i,j) from two CRX gates inside one 3-qubit space.
__device__ __forceinline__ float2 crx2_entry(int i, int j,
                                             int pc1, int pt1, float cA, float sA,
                                             int pc2, int pt2, float cB, float sB) {
    float2 sum = make_float2(0.f, 0.f);
#pragma unroll
    for (int k = 0; k < 8; ++k) {
        float2 g2 = crx_entry(i, k, pc2, pt2, cB, sB);
        float2 g1 = crx_entry(k, j, pc1, pt1, cA, sA);
        sum.x += g2.x * g1.x - g2.y * g1.y;
        sum.y += g2.x * g1.y + g2.y * g1.x;
    }
    return sum;
}

// Apply 8x8 complex unitary M on qubits (qh<qm<ql) to the full LDS state.
// Real 16x16 embedding R of M, GEMM D = R * W with W = 16 real comps x 512
// rest-configs: 32 16-column tiles (4 per wave), K=16 via four chained
// V_WMMA_F32_16X16X4_F32. Tiles are wave-disjoint -> no sync inside.
__device__ __forceinline__ void apply3(float2* __restrict__ st,
                                       const float2* __restrict__ M,
                                       int qh, int qm, int ql) {
    const int tid  = threadIdx.x;
    const int lane = tid & 31;
    const int wave = tid >> 5;
    const int col  = lane & 15;     // N within tile
    const int half = lane >> 4;     // 0: K 0-1 / rows 0-7, 1: K 2-3 / rows 8-15
    const int sh = 1 << (11 - qh);
    const int sm = 1 << (11 - qm);
    const int sl = 1 << (11 - ql);
    const unsigned tmask = (unsigned)(sh | sm | sl);

#if defined(__gfx1250__) && __has_builtin(__builtin_amdgcn_wmma_f32_16x16x4_f32)
    // A fragments: one complex entry of M per k-step, sign-twiddled per the
    // complex->real embedding (R[2i,2j]=Re, R[2i,2j+1]=-Im,
    // R[2i+1,2j]=Im, R[2i+1,2j+1]=Re).
    const int irow = col >> 1;      // complex row index i of M
    const int sgn  = col & 1;       // 0: real output row, 1: imag output row
    v2f afrag[4];
#pragma unroll
    for (int kk = 0; kk < 4; ++kk) {
        float2 mij = M[irow * 8 + (2 * kk + half)];
        afrag[kk].x = sgn ? mij.y : mij.x;
        afrag[kk].y = sgn ? mij.x : -mij.y;
    }
#pragma unroll
    for (int t = 0; t < 4; ++t) {
        int rest = (wave * 4 + t) * 16 + col;
        int base = 0, rb = rest;
#pragma unroll
        for (int bp = 0; bp < 12; ++bp) {      // scatter 9 rest bits
            if (!(tmask & (1u << bp))) { base |= (rb & 1) << bp; rb >>= 1; }
        }
        v8f acc = {0.f, 0.f, 0.f, 0.f, 0.f, 0.f, 0.f, 0.f};
#pragma unroll
        for (int kk = 0; kk < 4; ++kk) {
            int aidx = 2 * kk + half;          // amplitude index within group
            int off = ((aidx >> 2) & 1) * sh + ((aidx >> 1) & 1) * sm +
                      (aidx & 1) * sl;
            float2 amp = st[base + off];       // one ds_load_b64 per lane
            v2f bfrag; bfrag.x = amp.x; bfrag.y = amp.y;
            acc = __builtin_amdgcn_wmma_f32_16x16x4_f32(
                false, afrag[kk], false, bfrag, (short)0, acc, false, false);
        }
#pragma unroll
        for (int u = 0; u < 4; ++u) {          // D: 4 float2 amplitudes/lane
            int aidx = half * 4 + u;
            int off = ((aidx >> 2) & 1) * sh + ((aidx >> 1) & 1) * sm +
                      (aidx & 1) * sl;
            st[base + off] = make_float2(acc[2 * u], acc[2 * u + 1]);
        }
    }
#else
    // Scalar fallback (host pass / builtin unavailable): 8x8 complex matvec.
    for (int r = tid; r < 512; r += NTHREAD) {
        int base = 0, rb = r;
        for (int bp = 0; bp < 12; ++bp) {
            if (!(tmask & (1u << bp))) { base |= (rb & 1) << bp; rb >>= 1; }
        }
        float2 amps[8];
        for (int k = 0; k < 8; ++k) {
            int off = ((k >> 2) & 1) * sh + ((k >> 1) & 1) * sm + (k & 1) * sl;
            amps[k] = st[base + off];
        }
        for (int i = 0; i < 8; ++i) {
            float2 sum = make_float2(0.f, 0.f);
            for (int k = 0; k < 8; ++k) {
                float2 p = cmulf(M[i * 8 + k], amps[k]);
                sum.x += p.x; sum.y += p.y;
            }
            int off = ((i >> 2) & 1) * sh + ((i >> 1) & 1) * sm + (i & 1) * sl;
            st[base + off] = sum;
        }
    }
#endif
}

__global__ __launch_bounds__(NTHREAD)
void qfe_kernel(const float* __restrict__ params,
                const float* __restrict__ inputs,
                float* __restrict__ out) {
    __shared__ float2 st[DIM];            // 32 KB statevector
    __shared__ float2 Mall[NMAT * 64];    // 8 KB: all 8x8 unitaries of a phase
    __shared__ float2 Uq[4 * NQ];         // per-qubit fused 2x2 unitaries
    __shared__ float  pbuf[NPARAM];
    __shared__ float  ibuf[NQ];
    __shared__ float  red[3 * NQ];

    const int b   = blockIdx.x;
    const int tid = threadIdx.x;

    for (int i = tid; i < NPARAM; i += NTHREAD) pbuf[i] = params[b * NPARAM + i];
    if (tid < NQ) ibuf[tid] = inputs[b * NQ + tid];
    for (int i = tid; i < DIM; i += NTHREAD)
        st[i] = (i == 0) ? make_float2(1.f, 0.f) : make_float2(0.f, 0.f);
    __syncthreads();

    // ---- RY input embedding: 4 Kronecker unitaries, then 4 applications ----
    if (tid < NQ) {
        float c = __cosf(0.5f * ibuf[tid]), s = __sinf(0.5f * ibuf[tid]);
        Uq[4 * tid + 0] = make_float2(c, 0.f);
        Uq[4 * tid + 1] = make_float2(-s, 0.f);
        Uq[4 * tid + 2] = make_float2(s, 0.f);
        Uq[4 * tid + 3] = make_float2(c, 0.f);
    }
    __syncthreads();
    {   // 4 matrices * 64 entries = 256 -> one entry per thread
        int mat = tid >> 6, e = tid & 63;
        Mall[tid] = kron_entry(Uq, 3 * mat, 3 * mat + 1, 3 * mat + 2,
                               e >> 3, e & 7);
    }
    __syncthreads();
    for (int g = 0; g < 4; ++g) {
        apply3(st, Mall + 64 * g, 3 * g, 3 * g + 1, 3 * g + 2);
        __syncthreads();
    }

    // ---- layers ----
    for (int l = 0; l < 4; ++l) {
        const int base = l * 60;
        // fused U = RZ * RY * RX per qubit
        if (tid < NQ) {
            float t1 = pbuf[base + 3 * tid + 0];
            float t2 = pbuf[base + 3 * tid + 1];
            float t3 = pbuf[base + 3 * tid + 2];
            float c1 = __cosf(0.5f * t1), s1 = __sinf(0.5f * t1);
            float c2 = __cosf(0.5f * t2), s2 = __sinf(0.5f * t2);
            float c3 = __cosf(0.5f * t3), s3 = __sinf(0.5f * t3);
            float2 T00 = make_float2(c2 * c1,  s2 * s1);
            float2 T01 = make_float2(-s2 * c1, -c2 * s1);
            float2 T10 = make_float2(s2 * c1,  -c2 * s1);
            float2 T11 = make_float2(c2 * c1,  -s2 * s1);
            float2 e0 = make_float2(c3, -s3), e1 = make_float2(c3, s3);
            Uq[4 * tid + 0] = cmulf(e0, T00);
            Uq[4 * tid + 1] = cmulf(e0, T01);
            Uq[4 * tid + 2] = cmulf(e1, T10);
            Uq[4 * tid + 3] = cmulf(e1, T11);
        }
        __syncthreads();
        // Build all 16 unitaries of this layer (params-only, state-independent)
        for (int idx = tid; idx < NMAT * 64; idx += NTHREAD) {
            int mat = idx >> 6, e = idx & 63;
            int i = e >> 3, j = e & 7;
            float2 val;
            if (mat < 4) {
                val = kron_entry(Uq, 3 * mat, 3 * mat + 1, 3 * mat + 2, i, j);
            } else {
                int bwd = (mat >= 10);
                int m = mat - (bwd ? 10 : 4);
                const int* pp = bwd ? c_bwd_p[m] : c_fwd_p[m];
                float thA = pbuf[base + (bwd ? 48 : 36) + 2 * m];
                float thB = pbuf[base + (bwd ? 48 : 36) + 2 * m + 1];
                float cA = __cosf(0.5f * thA), sA = __sinf(0.5f * thA);
                float cB = __cosf(0.5f * thB), sB = __sinf(0.5f * thB);
                val = crx2_entry(i, j, pp[0], pp[1], cA, sA,
                                       pp[2], pp[3], cB, sB);
            }
            Mall[idx] = val;
        }
        __syncthreads();
        // Apply in circuit order: rotations, forward ring, backward ring.
        for (int g = 0; g < 4; ++g) {
            apply3(st, Mall + 64 * g, 3 * g, 3 * g + 1, 3 * g + 2);
            __syncthreads();
        }
        for (int m = 0; m < 6; ++m) {
            apply3(st, Mall + 64 * (4 + m),
                   c_fwd_q[m][0], c_fwd_q[m][1], c_fwd_q[m][2]);
            __syncthreads();
        }
        for (int m = 0; m < 6; ++m) {
            apply3(st, Mall + 64 * (10 + m),
                   c_bwd_q[m][0], c_bwd_q[m][1], c_bwd_q[m][2]);
            __syncthreads();
        }
    }

    // ---- expectation values: X, Y, Z per qubit ----
    if (tid < 3 * NQ) red[tid] = 0.f;
    __syncthreads();
    for (int q = 0; q < NQ; ++q) {
        int s = 1 << (11 - q);
        float xr = 0.f, yi = 0.f, zz = 0.f;
        for (int p = tid; p < 2048; p += NTHREAD) {
            int idx0 = ((p & ~(s - 1)) << 1) | (p & (s - 1));
            float2 a0 = st[idx0];
            float2 a1 = st[idx0 + s];
            xr += a0.x * a1.x + a0.y * a1.y;           // Re(conj(s0)*s1)
            yi += a0.x * a1.y - a0.y * a1.x;           // Im(conj(s0)*s1)
            zz += (a0.x * a0.x + a0.y * a0.y) - (a1.x * a1.x + a1.y * a1.y);
        }
#pragma unroll
        for (int o = 16; o > 0; o >>= 1) {             // wave32 reduction
            xr += __shfl_down(xr, o, 32);
            yi += __shfl_down(yi, o, 32);
            zz += __shfl_down(zz, o, 32);
        }
        if ((tid & 31) == 0) {
            atomicAdd(&red[q],          2.f * xr);
            atomicAdd(&red[NQ + q],     2.f * yi);
            atomicAdd(&red[2 * NQ + q], zz);
        }
    }
    __syncthreads();
    if (tid < 3 * NQ) out[b * (3 * NQ) + tid] = red[tid];
}

extern "C" void kernel_launch(void* const* d_in, const int* in_sizes, int n_in,
                              void* d_out, int out_size, void* d_ws,
                              size_t ws_size, hipStream_t stream) {
    (void)n_in; (void)out_size; (void)d_ws; (void)ws_size;
    const float* params = (const float*)d_in[0];
    const float* inputs = (const float*)d_in[1];
    float* out = (float*)d_out;
    const int batch = in_sizes[0] / NPARAM;            // 1024
    qfe_kernel<<<batch, NTHREAD, 0, stream>>>(params, inputs, out);
}